// IFFTConv_25881472926130
// MI455X (gfx1250) — compile-verified
//
#include <hip/hip_runtime.h>
#include <cstdint>

// ---------------------------------------------------------------------------
// ISTFT = (M=32768, K=1024, N=1024) f32 GEMM via V_WMMA_F32_16X16X4_F32
//         with async global->LDS (ASYNCcnt) double-buffered pipeline,
//         + gather-style overlap-add / window-normalize.
// Workspace requirement: frames buffer = 32768*1024*4 B = 128 MiB in d_ws.
// ---------------------------------------------------------------------------

typedef float v2f __attribute__((ext_vector_type(2)));
typedef float v8f __attribute__((ext_vector_type(8)));
typedef int   v4i __attribute__((vector_size(4 * sizeof(int))));

typedef __attribute__((address_space(1))) v4i* as1_v4i;
typedef __attribute__((address_space(3))) v4i* as3_v4i;
#define GPTR(p) ((as1_v4i)(uintptr_t)(p))
#define LPTR(p) ((as3_v4i)(uintptr_t)(p))

// global_load_async_to_lds_b128: per-lane 16B copy, memory -> LDS, no VGPR data
#if __has_builtin(__builtin_amdgcn_global_load_async_to_lds_b128)
#define ASYNC_CP16(gsrc, ldst) \
    __builtin_amdgcn_global_load_async_to_lds_b128(GPTR(gsrc), LPTR(ldst), 0, 0)
#else
#define ASYNC_CP16(gsrc, ldst)                                          \
    asm volatile("global_load_async_to_lds_b128 %0, %1, off"            \
                 :: "v"((uint32_t)(uintptr_t)(ldst)),                   \
                    "v"((uint64_t)(uintptr_t)(gsrc))                    \
                 : "memory")
#endif

#if __has_builtin(__builtin_amdgcn_s_wait_asynccnt)
#define WAIT_ASYNC0() __builtin_amdgcn_s_wait_asynccnt(0)
#else
#define WAIT_ASYNC0() asm volatile("s_wait_asynccnt 0x0" ::: "memory")
#endif

#define N_FFT    1024
#define HALF     512
#define HOP      256
#define AUDIO    524288
#define T_FRAMES 2048
#define BATCH    16
#define M_TOTAL  (BATCH * T_FRAMES)   // 32768
#define KDIM     1024
#define NDIM     1024

#define BM  128
#define BN  128
#define BK  32
#define LDA 36    // padded LDS stride for A (floats): 144B rows (16B aligned)
#define LDB 136   // padded LDS stride for B (floats): 544B rows; halves hit disjoint banks

// ---- async global -> LDS staging of one (128x32 A, 32x128 B) K-chunk ------
__device__ __forceinline__ void async_stage(const float* __restrict__ x,
                                            const float* __restrict__ W,
                                            int blockM0, int blockN0, int kb,
                                            int tid, float* As, float* Bs)
{
#pragma unroll
    for (int i = 0; i < 4; ++i) {
        int idx = tid + i * 256;          // 0..1023 float4 slots of the 128x32 tile
        int row = idx >> 3;               // 0..127
        int c4  = idx & 7;                // 0..7
        int m   = blockM0 + row;
        int b   = m >> 11;                // /2048
        int t   = m & (T_FRAMES - 1);
        int c   = kb + c4 * 4;            // BK multiple of 32 -> never splits channels
        int ch  = c >> 9;
        int cl  = c & (HALF - 1);
        const float* src = x + (size_t)(((b * 2 + ch) * T_FRAMES + t)) * HALF + cl;
        ASYNC_CP16(src, &As[row * LDA + c4 * 4]);
    }
#pragma unroll
    for (int i = 0; i < 4; ++i) {
        int idx = tid + i * 256;          // 0..1023 float4 slots of the 32x128 tile
        int row = idx >> 5;               // 0..31
        int n4  = idx & 31;               // 0..31
        const float* src = W + (size_t)(kb + row) * NDIM + blockN0 + n4 * 4;
        ASYNC_CP16(src, &Bs[row * LDB + n4 * 4]);
    }
}

// frames[m][k] = sum_c xc[m][c] * W[c][k]
// xc[m][c]: m = b*2048 + t ; c < 512 -> x[b][0][t][c], else x[b][1][t][c-512]
__global__ __launch_bounds__(256) void istft_gemm(const float* __restrict__ x,
                                                  const float* __restrict__ W,
                                                  float* __restrict__ frames)
{
    __shared__ float As[2][BM * LDA];
    __shared__ float Bs[2][BK * LDB];

    const int tid   = threadIdx.x;
    const int lane  = tid & 31;
    const int wave  = tid >> 5;     // 0..7
    const int waveM = wave >> 2;    // 0..1 -> 64-row strip
    const int waveN = wave & 3;     // 0..3 -> 32-col strip
    const int half  = lane >> 4;    // hi/lo half of wave
    const int l16   = lane & 15;

    const int blockN0 = blockIdx.x * BN;
    const int blockM0 = blockIdx.y * BM;

    v8f acc[4][2];
#pragma unroll
    for (int i = 0; i < 4; ++i)
#pragma unroll
        for (int j = 0; j < 2; ++j)
            acc[i][j] = (v8f){};

    // ---- prologue: async-stage chunk 0 into buffer 0
    async_stage(x, W, blockM0, blockN0, 0, tid, As[0], Bs[0]);
    WAIT_ASYNC0();
    __syncthreads();

    int cur = 0;
    for (int kb = 0; kb < KDIM; kb += BK, cur ^= 1) {
        const int nxt = cur ^ 1;
        const bool has_next = (kb + BK) < KDIM;

        // fire next chunk's async copies NOW; they land while we do 64 WMMAs
        if (has_next)
            async_stage(x, W, blockM0, blockN0, kb + BK, tid, As[nxt], Bs[nxt]);

        const float* Ac = As[cur];
        const float* Bc = Bs[cur];

        // ---- 8 k-steps of 4; 8 wmma per step; A frags reused 2x, B frags 4x
#pragma unroll
        for (int kk = 0; kk < BK; kk += 4) {
            // A 16x4 frag: VGPR0 = {K0 | K2}, VGPR1 = {K1 | K3} across lane halves
            v2f afrag[4];
#pragma unroll
            for (int i = 0; i < 4; ++i) {
                int row = waveM * 64 + i * 16 + l16;
                afrag[i] = *(const v2f*)&Ac[row * LDA + kk + half * 2];
            }
            // B 4x16 frag: N striped across lanes, K split across halves
            v2f bfrag[2];
#pragma unroll
            for (int j = 0; j < 2; ++j) {
                int n = waveN * 32 + j * 16 + l16;
                int r = kk + half * 2;
                bfrag[j].x = Bc[r * LDB + n];
                bfrag[j].y = Bc[(r + 1) * LDB + n];
            }
#pragma unroll
            for (int i = 0; i < 4; ++i)
#pragma unroll
                for (int j = 0; j < 2; ++j)
                    acc[i][j] = __builtin_amdgcn_wmma_f32_16x16x4_f32(
                        false, afrag[i], false, bfrag[j],
                        (short)0, acc[i][j], false, false);
        }

        // all of this wave's async copies must have landed before anyone reads nxt
        WAIT_ASYNC0();
        __syncthreads();
    }

    // ---- write 16x16 C tiles: VGPR r -> row (half*8 + r), col l16
    const int m0 = blockM0 + waveM * 64;
    const int n0 = blockN0 + waveN * 32;
#pragma unroll
    for (int i = 0; i < 4; ++i)
#pragma unroll
        for (int j = 0; j < 2; ++j)
#pragma unroll
            for (int r = 0; r < 8; ++r) {
                int row = m0 + i * 16 + half * 8 + r;
                int col = n0 + j * 16 + l16;
                frames[(size_t)row * NDIM + col] = acc[i][j][r];
            }
}

// out[b][s] = 2 * sum_t frames[b][t][p - 256 t] / (scale[s] + 1e-7), p = s + 384
__global__ __launch_bounds__(256) void istft_overlap(const float* __restrict__ frames,
                                                     const float* __restrict__ scale,
                                                     float* __restrict__ out)
{
    int gid = blockIdx.x * 256 + threadIdx.x;
    int b   = gid >> 19;                 // / AUDIO (2^19)
    int s   = gid & (AUDIO - 1);
    int p   = s + 384;                   // left pad of the cropped overlap-add buffer
    int tmin = (p - (N_FFT - HOP)) >> 8; // ceil((p-1023)/256) == floor((p-768)/256)
    if (tmin < 0) tmin = 0;
    int tmax = p >> 8;
    if (tmax > T_FRAMES - 1) tmax = T_FRAMES - 1;

    const float* fb = frames + (size_t)b * T_FRAMES * N_FFT;
    float acc = 0.0f;
    for (int t = tmin; t <= tmax; ++t)
        acc += fb[(size_t)t * N_FFT + (p - (t << 8))];

    out[gid] = 2.0f * acc / (scale[s] + 1e-7f);
}

extern "C" void kernel_launch(void* const* d_in, const int* in_sizes, int n_in,
                              void* d_out, int out_size, void* d_ws, size_t ws_size,
                              hipStream_t stream)
{
    const float* x     = (const float*)d_in[0];  // (16,2,2048,512) f32
    const float* W     = (const float*)d_in[1];  // (1024,1024) f32
    const float* scale = (const float*)d_in[2];  // (524288,) f32
    float* frames = (float*)d_ws;                // 128 MiB scratch
    float* out    = (float*)d_out;               // (16,524288) f32

    dim3 g1(NDIM / BN, M_TOTAL / BM);            // (8, 256) blocks
    istft_gemm<<<g1, 256, 0, stream>>>(x, W, frames);

    int total = BATCH * AUDIO;                   // 8,388,608
    istft_overlap<<<total / 256, 256, 0, stream>>>(frames, scale, out);
}